// MeshVoxelization_32057635897981
// MI455X (gfx1250) — compile-verified
//
#include <hip/hip_runtime.h>

typedef __attribute__((ext_vector_type(2))) float v2f;
typedef __attribute__((ext_vector_type(8))) float v8f;

#define VOL_R  256
#define NSAMP  6
#define KSAMP  36   // NSAMP^2 samples per face

// ---------------------------------------------------------------------------
// Kernel 1: clear the 256^3 occupancy grid (64 MB). float4 stores, one per thread.
// Output stays resident in the 192MB L2, so the scatter kernel hits L2.
// ---------------------------------------------------------------------------
__global__ void vox_zero_kernel(float4* __restrict__ out, int n4) {
    int i = blockIdx.x * blockDim.x + threadIdx.x;
    if (i < n4) out[i] = make_float4(0.f, 0.f, 0.f, 0.f);
}

// ---------------------------------------------------------------------------
// Kernel 2: WMMA voxelizer. Each wave handles 16 faces.
//   B (4x16, f32): col n = face n, rows = {vert0, vert1, vert2, 0} coord c
//   A (16x4, f32): row m = sample m of chunk, cols = {a, b, c, 0} weights
//   D = A x B  ->  D[m][n] = interpolated coordinate c of sample m on face n
// 3 sample chunks (16+16+4 of 36) x 3 coords = 9 v_wmma_f32_16x16x4_f32.
// D layout gives each lane x/y/z of the same samples -> direct scatter store.
// Scatter of constant 1.0 is idempotent -> plain stores, no atomics.
// ---------------------------------------------------------------------------
__global__ __launch_bounds__(256) void vox_wmma_kernel(
    const float* __restrict__ vertices,  // (V,3)
    const int*   __restrict__ faces,     // (F,3)
    float*       __restrict__ occ,       // (z,y,x) flattened 256^3
    int nfaces) {

    const int  lane = threadIdx.x & 31;
    const bool lo   = lane < 16;
    const int  mrow = lane & 15;
    const int  wave = (blockIdx.x * blockDim.x + threadIdx.x) >> 5;

    const int  f      = wave * 16 + mrow;   // this lane's B column (face)
    const bool fvalid = f < nfaces;

    // ---- Build B (4x16) per coordinate.
    // VGPR layout: v0 = rows {K=0 (lanes0-15), K=2 (lanes16-31)}, v1 = rows {K=1, K=3}.
    v2f Bx = {0.f, 0.f}, By = {0.f, 0.f}, Bz = {0.f, 0.f};
    if (fvalid) {
        const int k0  = lo ? 0 : 2;
        const int vi0 = faces[f * 3 + k0];
        Bx.x = vertices[vi0 * 3 + 0];
        By.x = vertices[vi0 * 3 + 1];
        Bz.x = vertices[vi0 * 3 + 2];
        if (lo) {  // row K=1 (vertex 1); lanes>=16 keep row K=3 == 0 (pad)
            const int vi1 = faces[f * 3 + 1];
            Bx.y = vertices[vi1 * 3 + 0];
            By.y = vertices[vi1 * 3 + 1];
            Bz.y = vertices[vi1 * 3 + 2];
        }
    }

    const v8f zero = {};

#pragma unroll
    for (int chunk = 0; chunk < 3; ++chunk) {
        // ---- Build A (16x4): barycentric weights for samples [chunk*16, chunk*16+16).
        // lanes0-15 carry (K=0,K=1)=(a,b) of row=lane; lanes16-31 carry (K=2,K=3)=(c,0).
        const int srow = chunk * 16 + mrow;
        float wa = 0.f, wb = 0.f, wc = 0.f;
        if (srow < KSAMP) {
            float aa = ((float)(srow / NSAMP) + 0.5f) * (1.0f / (float)NSAMP);
            float bb = ((float)(srow % NSAMP) + 0.5f) * (1.0f / (float)NSAMP);
            if (aa + bb > 1.0f) { aa = 1.0f - aa; bb = 1.0f - bb; }
            wa = aa; wb = bb; wc = 1.0f - aa - bb;
        }
        v2f A;
        A.x = lo ? wa : wc;
        A.y = lo ? wb : 0.0f;

        v8f Dx, Dy, Dz;
#if __has_builtin(__builtin_amdgcn_wmma_f32_16x16x4_f32)
        // EXEC is all-ones here (no divergence at this point), as WMMA requires.
        Dx = __builtin_amdgcn_wmma_f32_16x16x4_f32(false, A, false, Bx, (short)0, zero, false, false);
        Dy = __builtin_amdgcn_wmma_f32_16x16x4_f32(false, A, false, By, (short)0, zero, false, false);
        Dz = __builtin_amdgcn_wmma_f32_16x16x4_f32(false, A, false, Bz, (short)0, zero, false, false);
#else
        // Shuffle-based fallback replicating the WMMA data layout (compile safety net).
        float b0x = __shfl(Bx.x, mrow, 32),      b1x = __shfl(Bx.y, mrow, 32),      b2x = __shfl(Bx.x, mrow + 16, 32);
        float b0y = __shfl(By.x, mrow, 32),      b1y = __shfl(By.y, mrow, 32),      b2y = __shfl(By.x, mrow + 16, 32);
        float b0z = __shfl(Bz.x, mrow, 32),      b1z = __shfl(Bz.y, mrow, 32),      b2z = __shfl(Bz.x, mrow + 16, 32);
#pragma unroll
        for (int j = 0; j < 8; ++j) {
            int m = j + (lo ? 0 : 8);
            float am = __shfl(A.x, m, 32), bm = __shfl(A.y, m, 32), cm = __shfl(A.x, m + 16, 32);
            Dx[j] = am * b0x + bm * b1x + cm * b2x;
            Dy[j] = am * b0y + bm * b1y + cm * b2y;
            Dz[j] = am * b0z + bm * b1z + cm * b2z;
        }
#endif

        // ---- Scatter: lane holds D rows M=j (lanes0-15) / M=j+8 (lanes16-31), col N=mrow.
#pragma unroll
        for (int j = 0; j < 8; ++j) {
            const int s = chunk * 16 + j + (lo ? 0 : 8);
            if (fvalid && s < KSAMP) {
                const float px = Dx[j], py = Dy[j], pz = Dz[j];
                // idx = floor(p / (2/256) + 128) = floor(p*128 + 128), clamp [0,255]
                int ix = (int)floorf(px * 128.0f + 128.0f);
                int iy = (int)floorf(py * 128.0f + 128.0f);
                int iz = (int)floorf(pz * 128.0f + 128.0f);
                ix = min(max(ix, 0), VOL_R - 1);
                iy = min(max(iy, 0), VOL_R - 1);
                iz = min(max(iz, 0), VOL_R - 1);
                // output layout (z,y,x):  occ[z][y][x] = 1
                occ[(((iz << 8) + iy) << 8) + ix] = 1.0f;
            }
        }
    }
}

// ---------------------------------------------------------------------------
extern "C" void kernel_launch(void* const* d_in, const int* in_sizes, int n_in,
                              void* d_out, int out_size, void* d_ws, size_t ws_size,
                              hipStream_t stream) {
    const float* vertices = (const float*)d_in[0];   // (100000, 3) f32
    const int*   faces    = (const int*)d_in[1];     // (200000, 3) i32
    float*       out      = (float*)d_out;           // 256^3 f32, (z,y,x)

    const int nfaces = in_sizes[1] / 3;

    // 1) clear grid (re-initializes d_out every call; deterministic)
    const int n4 = out_size / 4;   // 16777216 / 4 = 4194304 float4 stores
    vox_zero_kernel<<<(n4 + 255) / 256, 256, 0, stream>>>((float4*)out, n4);

    // 2) voxelize: one wave per 16 faces, 8 waves (256 threads) per block
    const int waves   = (nfaces + 15) / 16;
    const int threads = waves * 32;
    const int blocks  = (threads + 255) / 256;
    vox_wmma_kernel<<<blocks, 256, 0, stream>>>(vertices, faces, out, nfaces);
}